// LevelLayer_54752243089463
// MI455X (gfx1250) — compile-verified
//
#include <hip/hip_runtime.h>
#include <hip/hip_bf16.h>

#define NN   8192
#define GG   32
#define PER  256
#define KNN  15
#define FIN  4
#define FOUT 4
#define LAT  64
#define HID  128

typedef __attribute__((ext_vector_type(2))) float v2f;
typedef __attribute__((ext_vector_type(8))) float v8f;

// Accumulate one 16x16 f32 tile: C += A(16xK, row-major, stride lda) * B(KxN slice, stride ldb).
// Per ISA 7.12.2 (wave32):
//   A 16x4 f32 -> 2 VGPRs: lanes 0-15 hold K=0 (v0), K=1 (v1); lanes 16-31 hold K=2,K=3.
//   B 4x16     -> 2 VGPRs: lanes 0-15 hold row K=0 (v0), K=1 (v1); lanes 16-31 rows K=2,K=3.
__device__ __forceinline__ v8f wmma_f32_acc(const float* __restrict__ A, int lda,
                                            const float* __restrict__ B, int ldb,
                                            int K, int lane, v8f c) {
    const int l16  = lane & 15;
    const int half = lane >> 4;
    #pragma unroll 4
    for (int k0 = 0; k0 < K; k0 += 4) {
        const int ka = k0 + 2 * half;
        v2f a, b;
        a.x = A[l16 * lda + ka];
        a.y = A[l16 * lda + ka + 1];
        b.x = B[ka * ldb + l16];
        b.y = B[(ka + 1) * ldb + l16];
        c = __builtin_amdgcn_wmma_f32_16x16x4_f32(false, a, false, b, (short)0, c,
                                                  false, false);
    }
    return c;
}

// ---------------- space_emb: h = leaky(x@W1+b1)@W2+b2 ----------------
__global__ void k_space_emb(const float* __restrict__ x,
                            const float* __restrict__ W1, const float* __restrict__ b1,
                            const float* __restrict__ W2, const float* __restrict__ b2,
                            float* __restrict__ hout) {
    __shared__ float t1[64 * 68];          // 64 rows, stride 68 (bank rotation)
    const int tid  = threadIdx.x;
    const int base = blockIdx.x * 64;

    // Layer 1 (K=4): VALU, one (row, 16-col group) per thread
    {
        const int row  = tid >> 2;
        const int q    = tid & 3;
        const int node = base + row;
        float xf[FIN];
        #pragma unroll
        for (int f = 0; f < FIN; ++f) xf[f] = x[node * FIN + f];
        #pragma unroll
        for (int cc = 0; cc < 16; ++cc) {
            const int c = q * 16 + cc;
            float acc = b1[c];
            #pragma unroll
            for (int f = 0; f < FIN; ++f) acc += xf[f] * W1[f * LAT + c];
            t1[row * 68 + c] = acc > 0.f ? acc : 0.01f * acc;
        }
    }
    __syncthreads();

    // Layer 2 (64x64x64): WMMA, 16 tiles over 8 waves
    const int wave = tid >> 5, lane = tid & 31;
    const int l16 = lane & 15, half = lane >> 4;
    const int m0 = (wave >> 1) * 16;
    #pragma unroll
    for (int t = 0; t < 2; ++t) {
        const int n0 = ((wave & 1) * 2 + t) * 16;
        v8f c;
        const float bias = b2[n0 + l16];
        #pragma unroll
        for (int r = 0; r < 8; ++r) c[r] = bias;
        c = wmma_f32_acc(&t1[m0 * 68], 68, W2 + n0, LAT, LAT, lane, c);
        #pragma unroll
        for (int r = 0; r < 8; ++r) {
            const int row = m0 + r + 8 * half;
            hout[(base + row) * LAT + n0 + l16] = c[r];
        }
    }
}

// ---------------- knn: per-event top-15 on first 3 dims of h ----------------
__global__ void k_knn(const float* __restrict__ h, int* __restrict__ nbr,
                      int* __restrict__ ei0, int* __restrict__ ei1) {
    __shared__ float pos[PER][4];
    const int i = threadIdx.x;
    const int base = blockIdx.x * PER;
    pos[i][0] = h[(base + i) * LAT + 0];
    pos[i][1] = h[(base + i) * LAT + 1];
    pos[i][2] = h[(base + i) * LAT + 2];
    __syncthreads();

    float bd[KNN]; int bj[KNN];
    #pragma unroll
    for (int k = 0; k < KNN; ++k) { bd[k] = 3.0e38f; bj[k] = 0; }
    const float px = pos[i][0], py = pos[i][1], pz = pos[i][2];
    for (int j = 0; j < PER; ++j) {
        if (j == i) continue;
        const float dx = px - pos[j][0];
        const float dy = py - pos[j][1];
        const float dz = pz - pos[j][2];
        const float d = dx * dx + dy * dy + dz * dz;
        if (d < bd[KNN - 1]) {               // insertion keeps ascending order (== top_k(-d))
            int p = KNN - 1;
            while (p > 0 && bd[p - 1] > d) {
                bd[p] = bd[p - 1]; bj[p] = bj[p - 1]; --p;
            }
            bd[p] = d; bj[p] = j;
        }
    }
    const int node = base + i;
    #pragma unroll
    for (int k = 0; k < KNN; ++k) {
        const int src = base + bj[k];
        nbr[node * KNN + k] = src;
        ei0[node * KNN + k] = src;           // edge_index[0] = neighbor (source)
        ei1[node * KNN + k] = node;          // edge_index[1] = center (target)
    }
}

// ---------------- GIN layer: hout = hin + MLP(hin + sum_nbr hin) ----------------
__global__ void k_gin(const float* __restrict__ hin, float* __restrict__ hout,
                      const int* __restrict__ nbr,
                      const float* __restrict__ Wa, const float* __restrict__ ba,
                      const float* __restrict__ Wb, const float* __restrict__ bb) {
    __shared__ float m[64 * 68];             // 17 KB
    __shared__ float t[64 * 132];            // 33 KB
    const int tid  = threadIdx.x;
    const int base = blockIdx.x * 64;

    // Stage 1: GIN aggregation m = h_i + sum_j h_j (neighbors hit L2/WGP$)
    {
        const int row = tid >> 2, q = tid & 3;
        const int node = base + row;
        float acc[16];
        #pragma unroll
        for (int cc = 0; cc < 16; ++cc) acc[cc] = hin[node * LAT + q * 16 + cc];
        for (int k = 0; k < KNN; ++k) {
            const int j = nbr[node * KNN + k];
            #pragma unroll
            for (int cc = 0; cc < 16; ++cc) acc[cc] += hin[j * LAT + q * 16 + cc];
        }
        #pragma unroll
        for (int cc = 0; cc < 16; ++cc) m[row * 68 + q * 16 + cc] = acc[cc];
    }
    // warm GL2/WGP$ for the second GEMM's weights while we aggregate
    __builtin_prefetch(Wb + ((tid * 16) & (HID * LAT - 1)), 0, 1);
    __syncthreads();

    const int wave = tid >> 5, lane = tid & 31;
    const int l16 = lane & 15, half = lane >> 4;
    const int m0 = (wave >> 1) * 16;

    // GEMM1: t = relu(m @ Wa + ba), 64x64x128 -> 32 tiles / 8 waves
    #pragma unroll
    for (int ti = 0; ti < 4; ++ti) {
        const int n0 = ((wave & 1) * 4 + ti) * 16;
        v8f c;
        const float bias = ba[n0 + l16];
        #pragma unroll
        for (int r = 0; r < 8; ++r) c[r] = bias;
        c = wmma_f32_acc(&m[m0 * 68], 68, Wa + n0, HID, LAT, lane, c);
        #pragma unroll
        for (int r = 0; r < 8; ++r) {
            const int row = m0 + r + 8 * half;
            const float v = c[r];
            t[row * 132 + n0 + l16] = v > 0.f ? v : 0.f;
        }
    }
    __syncthreads();

    // GEMM2: out = t @ Wb + bb, 64x128x64; fuse residual into the store
    #pragma unroll
    for (int ti = 0; ti < 2; ++ti) {
        const int n0 = ((wave & 1) * 2 + ti) * 16;
        v8f c;
        const float bias = bb[n0 + l16];
        #pragma unroll
        for (int r = 0; r < 8; ++r) c[r] = bias;
        c = wmma_f32_acc(&t[m0 * 132], 132, Wb + n0, LAT, HID, lane, c);
        #pragma unroll
        for (int r = 0; r < 8; ++r) {
            const int row = m0 + r + 8 * half;
            const int idx = (base + row) * LAT + n0 + l16;
            hout[idx] = hin[idx] + c[r];
        }
    }
}

// ---------------- out_emb: x_emb = leaky(h@W1+b1)@W2+b2 (FOUT=4) ----------------
__global__ void k_out_emb(const float* __restrict__ h,
                          const float* __restrict__ W1, const float* __restrict__ b1,
                          const float* __restrict__ W2, const float* __restrict__ b2,
                          float* __restrict__ xemb) {
    __shared__ float t[64 * 68];
    const int tid  = threadIdx.x;
    const int base = blockIdx.x * 64;
    const int wave = tid >> 5, lane = tid & 31;
    const int l16 = lane & 15, half = lane >> 4;

    // GEMM1: 64x64x64 WMMA, A straight from global h (L2-resident), leaky into LDS
    {
        const int m0 = (wave >> 1) * 16;
        #pragma unroll
        for (int ti = 0; ti < 2; ++ti) {
            const int n0 = ((wave & 1) * 2 + ti) * 16;
            v8f c;
            const float bias = b1[n0 + l16];
            #pragma unroll
            for (int r = 0; r < 8; ++r) c[r] = bias;
            c = wmma_f32_acc(h + (size_t)(base + m0) * LAT, LAT, W1 + n0, LAT, LAT, lane, c);
            #pragma unroll
            for (int r = 0; r < 8; ++r) {
                const int row = m0 + r + 8 * half;
                const float v = c[r];
                t[row * 68 + n0 + l16] = v > 0.f ? v : 0.01f * v;
            }
        }
    }
    __syncthreads();

    // Layer 2 (N=4): VALU, one (row, out-col) per thread
    {
        const int row = tid >> 2, c = tid & 3;
        float acc = b2[c];
        #pragma unroll
        for (int k = 0; k < LAT; ++k) acc += t[row * 68 + k] * W2[k * FOUT + c];
        xemb[(base + row) * FOUT + c] = acc;
    }
}

extern "C" void kernel_launch(void* const* d_in, const int* in_sizes, int n_in,
                              void* d_out, int out_size, void* d_ws, size_t ws_size,
                              hipStream_t stream) {
    (void)in_sizes; (void)n_in; (void)out_size; (void)ws_size;

    const float* x     = (const float*)d_in[0];
    // d_in[1] = batch (implied by event structure), d_in[2] = condition (unused, n_cond=0)
    const float* W_se1 = (const float*)d_in[3];
    const float* b_se1 = (const float*)d_in[4];
    const float* W_se2 = (const float*)d_in[5];
    const float* b_se2 = (const float*)d_in[6];
    const float* W_g1a = (const float*)d_in[7];
    const float* b_g1a = (const float*)d_in[8];
    const float* W_g1b = (const float*)d_in[9];
    const float* b_g1b = (const float*)d_in[10];
    const float* W_g2a = (const float*)d_in[11];
    const float* b_g2a = (const float*)d_in[12];
    const float* W_g2b = (const float*)d_in[13];
    const float* b_g2b = (const float*)d_in[14];
    const float* W_oe1 = (const float*)d_in[15];
    const float* b_oe1 = (const float*)d_in[16];
    const float* W_oe2 = (const float*)d_in[17];
    const float* b_oe2 = (const float*)d_in[18];

    // Output layout (tuple flattened): h [N*LAT] f32 | x_emb [N*FOUT] f32 | ei [2*N*K] i32 bits
    float* out      = (float*)d_out;
    float* h_out    = out;
    float* xemb_out = out + (size_t)NN * LAT;
    int*   ei0      = (int*)(out + (size_t)NN * LAT + (size_t)NN * FOUT);
    int*   ei1      = ei0 + (size_t)NN * KNN;

    // Workspace: hA, hB double buffers + neighbor table
    float* hA  = (float*)d_ws;
    float* hB  = hA + (size_t)NN * LAT;
    int*   nbr = (int*)(hB + (size_t)NN * LAT);

    k_space_emb<<<NN / 64, 256, 0, stream>>>(x, W_se1, b_se1, W_se2, b_se2, hA);
    k_knn<<<GG, PER, 0, stream>>>(hA, nbr, ei0, ei1);
    k_gin<<<NN / 64, 256, 0, stream>>>(hA, hB, nbr, W_g1a, b_g1a, W_g1b, b_g1b);
    k_gin<<<NN / 64, 256, 0, stream>>>(hB, h_out, nbr, W_g2a, b_g2a, W_g2b, b_g2b);
    k_out_emb<<<NN / 64, 256, 0, stream>>>(h_out, W_oe1, b_oe1, W_oe2, b_oe2, xemb_out);
}